// PScanTriton_21217138442551
// MI455X (gfx1250) — compile-verified
//
#include <hip/hip_runtime.h>
#include <stdint.h>

// ---------------------------------------------------------------------------
// Complex first-order scan  y_t = a_t * y_{t-1} + x_t   (B=8, L=4096, D=512)
//
// Memory-bound (≈670 MB total traffic for the 2-pass scheme → ~29 us at
// 23.3 TB/s). Strategy: 3-phase chunked associative scan, float4-vectorized
// per thread, with CDNA5 async global->LDS staging (double buffered,
// tracked by ASYNCcnt) to keep deep prefetch in flight per wave.
// ---------------------------------------------------------------------------

#ifndef __has_builtin
#define __has_builtin(x) 0
#endif

#define AS1 __attribute__((address_space(1)))
#define AS3 __attribute__((address_space(3)))

typedef int v4i __attribute__((vector_size(16)));   // int4 vector (16 B)

constexpr int B   = 8;
constexpr int L   = 4096;
constexpr int D   = 512;
constexpr int CH  = 32;        // chunks along L
constexpr int LC  = L / CH;    // 128 timesteps per chunk
constexpr int TS  = 2;         // timesteps per async tile
constexpr int NT  = LC / TS;   // 64 tiles per chunk
constexpr int D4  = D / 4;     // 128 float4 channel groups
constexpr int BLK = 256;       // 8 waves (wave32)
constexpr int NSEQ4 = B * CH * D4;  // 32768 threads in heavy phases
constexpr int BCD   = B * CH * D;   // 131072 floats per ws array

// LDS staging: [buf][array(AR,AI,XR,XI)][step][thread] of float4 = 64 KB
typedef float4 Smem[2][4][TS][BLK];

// ---- CDNA5 async global->LDS copy (ASYNCcnt-tracked) ----------------------
__device__ __forceinline__ void async_b128(const float* g, float4* l) {
#if __has_builtin(__builtin_amdgcn_global_load_async_to_lds_b128)
  __builtin_amdgcn_global_load_async_to_lds_b128(
      (AS1 v4i*)g, (AS3 v4i*)l, 0, 0);
#else
  unsigned lds_off = (unsigned)(uintptr_t)(AS3 void*)l;
  asm volatile("global_load_async_to_lds_b128 %0, %1, off"
               :: "v"(lds_off), "v"((unsigned long long)(uintptr_t)g)
               : "memory");
#endif
}

#if __has_builtin(__builtin_amdgcn_s_wait_asynccnt)
#define WAIT_ASYNC(n) __builtin_amdgcn_s_wait_asynccnt(n)
#else
#define WAIT_ASYNC(n) asm volatile("s_wait_asynccnt %0" :: "i"(n) : "memory")
#endif

__device__ __forceinline__ void issue_tile(Smem& sm, int buf,
    const float* Ar, const float* Ai, const float* Xr, const float* Xi,
    size_t gbase, int t) {
#pragma unroll
  for (int s = 0; s < TS; ++s) {
    const size_t gi = gbase + (size_t)s * D;
    async_b128(Ar + gi, &sm[buf][0][s][t]);
    async_b128(Ai + gi, &sm[buf][1][s][t]);
    async_b128(Xr + gi, &sm[buf][2][s][t]);
    async_b128(Xi + gi, &sm[buf][3][s][t]);
  }
}

// ---- Phase 1: per-chunk aggregates (A_prod, X_agg), 4 channels/thread -----
__global__ __launch_bounds__(BLK) void pscan_phase1_chunk_agg(
    const float* __restrict__ Ar, const float* __restrict__ Ai,
    const float* __restrict__ Xr, const float* __restrict__ Xi,
    float* __restrict__ cAr, float* __restrict__ cAi,
    float* __restrict__ cXr, float* __restrict__ cXi) {
  __shared__ Smem sm;
  const int t   = threadIdx.x;
  const int tid = blockIdx.x * BLK + t;
  const int d4  = tid % D4;
  const int c   = (tid / D4) % CH;
  const int b   = tid / (D4 * CH);
  const size_t base = ((size_t)b * L + (size_t)c * LC) * D + (size_t)d4 * 4;

  issue_tile(sm, 0, Ar, Ai, Xr, Xi, base, t);

  float4 aAr = make_float4(1.f, 1.f, 1.f, 1.f);
  float4 aAi = make_float4(0.f, 0.f, 0.f, 0.f);
  float4 aXr = make_float4(0.f, 0.f, 0.f, 0.f);
  float4 aXi = make_float4(0.f, 0.f, 0.f, 0.f);

  for (int tt = 0; tt < NT; ++tt) {
    const int buf = tt & 1;
    if (tt + 1 < NT) {
      issue_tile(sm, buf ^ 1, Ar, Ai, Xr, Xi,
                 base + (size_t)(tt + 1) * TS * D, t);
      WAIT_ASYNC(TS * 4);   // previous tile's 8 asyncs complete (in-order)
    } else {
      WAIT_ASYNC(0);
    }
    asm volatile("" ::: "memory");
#pragma unroll
    for (int s = 0; s < TS; ++s) {
      const float4 ar = sm[buf][0][s][t];
      const float4 ai = sm[buf][1][s][t];
      const float4 xr = sm[buf][2][s][t];
      const float4 xi = sm[buf][3][s][t];
      // agg := (a,x) o agg :  A' = a*A (cplx),  X' = a*X + x (cplx)
#define COMB(f) { const float tr = ar.f, ti = ai.f;                     \
      const float nAr = tr*aAr.f - ti*aAi.f;                            \
      const float nAi = tr*aAi.f + ti*aAr.f;                            \
      const float nXr = tr*aXr.f - ti*aXi.f + xr.f;                     \
      const float nXi = tr*aXi.f + ti*aXr.f + xi.f;                     \
      aAr.f = nAr; aAi.f = nAi; aXr.f = nXr; aXi.f = nXi; }
      COMB(x) COMB(y) COMB(z) COMB(w)
#undef COMB
    }
  }
  const size_t k = ((size_t)b * CH + c) * D + (size_t)d4 * 4;
  *(float4*)(cAr + k) = aAr;
  *(float4*)(cAi + k) = aAi;
  *(float4*)(cXr + k) = aXr;
  *(float4*)(cXi + k) = aXi;
}

// ---- Phase 2: scan the CH chunk aggregates -> exclusive carries -----------
// Carries overwrite cXr/cXi in place (aggregate X no longer needed after).
__global__ __launch_bounds__(BLK) void pscan_phase2_carry(
    const float* __restrict__ cAr, const float* __restrict__ cAi,
    float* __restrict__ cXr, float* __restrict__ cXi) {
  const int tid = blockIdx.x * BLK + threadIdx.x;  // 0 .. B*D4-1
  const int d4  = tid % D4;
  const int b   = tid / D4;
  float4 yr = make_float4(0.f, 0.f, 0.f, 0.f);
  float4 yi = make_float4(0.f, 0.f, 0.f, 0.f);
  for (int c = 0; c < CH; ++c) {
    const size_t k = ((size_t)b * CH + c) * D + (size_t)d4 * 4;
    const float4 ar = *(const float4*)(cAr + k);
    const float4 ai = *(const float4*)(cAi + k);
    const float4 xr = *(const float4*)(cXr + k);
    const float4 xi = *(const float4*)(cXi + k);
    *(float4*)(cXr + k) = yr;   // y at chunk entry (exclusive prefix)
    *(float4*)(cXi + k) = yi;
#define STEP(f) { const float nr = ar.f*yr.f - ai.f*yi.f + xr.f;        \
                  const float ni = ar.f*yi.f + ai.f*yr.f + xi.f;        \
                  yr.f = nr; yi.f = ni; }
    STEP(x) STEP(y) STEP(z) STEP(w)
#undef STEP
  }
}

// ---- Phase 3: re-stream inputs, apply recurrence from carry, write out ----
__global__ __launch_bounds__(BLK) void pscan_phase3_apply(
    const float* __restrict__ Ar, const float* __restrict__ Ai,
    const float* __restrict__ Xr, const float* __restrict__ Xi,
    const float* __restrict__ carR, const float* __restrict__ carI,
    float* __restrict__ out) {
  __shared__ Smem sm;
  const int t   = threadIdx.x;
  const int tid = blockIdx.x * BLK + t;
  const int d4  = tid % D4;
  const int c   = (tid / D4) % CH;
  const int b   = tid / (D4 * CH);
  const size_t base = ((size_t)b * L + (size_t)c * LC) * D + (size_t)d4 * 4;

  issue_tile(sm, 0, Ar, Ai, Xr, Xi, base, t);

  const size_t k = ((size_t)b * CH + c) * D + (size_t)d4 * 4;
  float4 yr = *(const float4*)(carR + k);
  float4 yi = *(const float4*)(carI + k);

  for (int tt = 0; tt < NT; ++tt) {
    const int buf = tt & 1;
    if (tt + 1 < NT) {
      issue_tile(sm, buf ^ 1, Ar, Ai, Xr, Xi,
                 base + (size_t)(tt + 1) * TS * D, t);
      WAIT_ASYNC(TS * 4);
    } else {
      WAIT_ASYNC(0);
    }
    asm volatile("" ::: "memory");
#pragma unroll
    for (int s = 0; s < TS; ++s) {
      const float4 ar = sm[buf][0][s][t];
      const float4 ai = sm[buf][1][s][t];
      const float4 xr = sm[buf][2][s][t];
      const float4 xi = sm[buf][3][s][t];
#define STEP(f) { const float nr = ar.f*yr.f - ai.f*yi.f + xr.f;        \
                  const float ni = ar.f*yi.f + ai.f*yr.f + xi.f;        \
                  yr.f = nr; yi.f = ni; }
      STEP(x) STEP(y) STEP(z) STEP(w)
#undef STEP
      // out[b, l, d..d+3, 0:2] -> 32 contiguous bytes per thread
      const size_t o = (base + (size_t)(tt * TS + s) * D) * 2;
      *(float4*)(out + o)     = make_float4(yr.x, yi.x, yr.y, yi.y);
      *(float4*)(out + o + 4) = make_float4(yr.z, yi.z, yr.w, yi.w);
    }
  }
}

// ---------------------------------------------------------------------------
extern "C" void kernel_launch(void* const* d_in, const int* in_sizes, int n_in,
                              void* d_out, int out_size, void* d_ws,
                              size_t ws_size, hipStream_t stream) {
  (void)in_sizes; (void)n_in; (void)out_size; (void)ws_size;
  const float* Ar = (const float*)d_in[0];
  const float* Ai = (const float*)d_in[1];
  const float* Xr = (const float*)d_in[2];
  const float* Xi = (const float*)d_in[3];
  float* out = (float*)d_out;

  // Workspace: 4 arrays of B*CH*D floats = 2 MB total.
  float* ws  = (float*)d_ws;
  float* cAr = ws;
  float* cAi = cAr + BCD;
  float* cXr = cAi + BCD;   // overwritten with carry(re) by phase 2
  float* cXi = cXr + BCD;   // overwritten with carry(im) by phase 2

  pscan_phase1_chunk_agg<<<NSEQ4 / BLK, BLK, 0, stream>>>(
      Ar, Ai, Xr, Xi, cAr, cAi, cXr, cXi);
  pscan_phase2_carry<<<(B * D4) / BLK, BLK, 0, stream>>>(
      cAr, cAi, cXr, cXi);
  pscan_phase3_apply<<<NSEQ4 / BLK, BLK, 0, stream>>>(
      Ar, Ai, Xr, Xi, cXr, cXi, out);
}